// VitsModel_78709570667084
// MI455X (gfx1250) — compile-verified
//
#include <hip/hip_runtime.h>
#include <hip/hip_bf16.h>

// ---------------------------------------------------------------------------
// VITS monotonic alignment search (MAS) pipeline for gfx1250 / MI455X.
//   B=32, C=192, Tx=512, Ty=2048
// Stage 1: prep A^T (o_scale, m_p*o_scale) + per-(b,x) bias reduction
// Stage 2: batched GEMM (K=384) via v_wmma_f32_16x16x4_f32, masked epilogue,
//          stored transposed as lp_t[B][Ty][Tx] (scratch inside attn output).
//          K-pair interleaved LDS (one ds_load_b64 per WMMA fragment) with
//          double buffering + register staging (one barrier per K chunk).
// Stage 3: forward Viterbi DP over y (ballot-packed direction bits)
// Stage 4: backtrack per batch with dirs plane cached in LDS (320KB/WGP)
// Stage 5: fill attn one-hot, durations, and gather-expanded m_p/logs_p
// ---------------------------------------------------------------------------

#define B_   32
#define C_   192
#define TX   512
#define TY   2048
#define K2   (2 * C_)          // 384
#define NEGV (-1e9f)

#define BTM  64                // y-tile per block (WMMA M)
#define BTN  128               // x-tile per block (WMMA N)
#define KC   16                // K staged per LDS chunk
#define KP   (KC / 2)          // K pairs per chunk
#define NCHUNK (K2 / KC)       // 24
#define BTM_P 80               // padded v2f row (stride 160 dw ≡ 32 mod 64)
#define BTN_P 144              // padded v2f row (stride 288 dw ≡ 32 mod 64)
#define HALF_LOG_2PI 0.91893853320467274178f

typedef float v2f __attribute__((ext_vector_type(2)));
typedef float v8f __attribute__((ext_vector_type(8)));

__device__ __forceinline__ v8f wmma4(v2f a, v2f b, v8f c) {
    // D = A(16x4,f32) * B(4x16,f32) + C(16x16,f32)
    return __builtin_amdgcn_wmma_f32_16x16x4_f32(
        false, a, false, b, (short)0, c, false, false);
}

// ---------------------------------------------------------------------------
// Stage 1a: A^T[b][k][x]:  k<192 -> exp(-2*logs), k>=192 -> m * exp(-2*logs)
// ---------------------------------------------------------------------------
__global__ __launch_bounds__(256) void prep_a_kernel(
    const float* __restrict__ mp, const float* __restrict__ ls,
    float* __restrict__ Ap)
{
    size_t i = (size_t)blockIdx.x * 256 + threadIdx.x;   // over B*C*Tx
    if (i >= (size_t)B_ * C_ * TX) return;
    int x = (int)(i % TX);
    size_t bc = i / TX;
    int c = (int)(bc % C_);
    int b = (int)(bc / C_);
    float l  = ls[i];
    float os = __expf(-2.0f * l);
    float m  = mp[i];
    float* ob = Ap + (size_t)b * K2 * TX;
    ob[(size_t)c * TX + x]        = os;
    ob[(size_t)(C_ + c) * TX + x] = m * os;
}

// Stage 1b: bias[b][x] = sum_c(-0.5*log(2pi) - logs) + sum_c(-0.5*m^2*os)
__global__ __launch_bounds__(256) void prep_bias_kernel(
    const float* __restrict__ mp, const float* __restrict__ ls,
    float* __restrict__ bias)
{
    int i = blockIdx.x * 256 + threadIdx.x;              // over B*Tx
    if (i >= B_ * TX) return;
    int x = i % TX, b = i / TX;
    const float* mb = mp + (size_t)b * C_ * TX + x;
    const float* lb = ls + (size_t)b * C_ * TX + x;
    float s = 0.0f;
    for (int c = 0; c < C_; ++c) {
        float l  = lb[(size_t)c * TX];
        float m  = mb[(size_t)c * TX];
        float os = __expf(-2.0f * l);
        s += -HALF_LOG_2PI - l - 0.5f * m * m * os;
    }
    bias[i] = s;
}

// ---------------------------------------------------------------------------
// Stage 2 GEMM helpers: register staging of one K chunk per thread.
// ---------------------------------------------------------------------------
struct ZStage { float2 r0, r1; };
struct MStage { float4 q0, q1; };

__device__ __forceinline__ ZStage load_z(const float* __restrict__ zb,
                                         int kc0, int kp, int ypos)
{
    int c = kc0 + 2 * kp;
    bool sq = (kc0 < C_);            // first 192 K rows carry -0.5*z^2
    if (!sq) c -= C_;
    ZStage s;
    s.r0 = *(const float2*)(zb + (size_t)c * TY + ypos);
    s.r1 = *(const float2*)(zb + (size_t)(c + 1) * TY + ypos);
    if (sq) {
        s.r0.x = -0.5f * s.r0.x * s.r0.x; s.r0.y = -0.5f * s.r0.y * s.r0.y;
        s.r1.x = -0.5f * s.r1.x * s.r1.x; s.r1.y = -0.5f * s.r1.y * s.r1.y;
    }
    return s;
}

__device__ __forceinline__ MStage load_m(const float* __restrict__ ab,
                                         int kc0, int kp, int xpos)
{
    int k = kc0 + 2 * kp;
    MStage s;
    s.q0 = *(const float4*)(ab + (size_t)k * TX + xpos);
    s.q1 = *(const float4*)(ab + (size_t)(k + 1) * TX + xpos);
    return s;
}

// ---------------------------------------------------------------------------
// Stage 2: lp_t[b][y][x] = sum_k Z'[k][y] * A^T[k][x] + bias[b][x], masked.
// WMMA mapping: M = y, N = x  -> D lanes contiguous in x -> coalesced stores.
// ---------------------------------------------------------------------------
__global__ __launch_bounds__(256) void gemm_logp_kernel(
    const float* __restrict__ zp, const float* __restrict__ Ap,
    const float* __restrict__ bias, const int* __restrict__ xlen,
    const int* __restrict__ ylen, float* __restrict__ lp_t)
{
    // K-pair interleaved: element = {val[2kp][col], val[2kp+1][col]}
    __shared__ v2f Zs[2][KP][BTM_P];
    __shared__ v2f Ms[2][KP][BTN_P];

    const int b   = blockIdx.z;
    const int y0  = blockIdx.y * BTM;
    const int x0  = blockIdx.x * BTN;
    const int tid = threadIdx.x;
    const int lane = tid & 31, wid = tid >> 5;
    const int wm = wid & 1, wn = wid >> 1;          // 2 x 4 wave grid
    const int half = lane >> 4, l16 = lane & 15;
    const int kb = half * 2;                        // frag K offset (0 or 2)

    // loader coordinates (all 256 threads participate in both loads)
    const int lkp  = tid >> 5;                      // 0..7  K pair
    const int ly   = (tid & 31) * 2;                // 0..62 (2 y per thread)
    const int lx   = (tid & 31) * 4;                // 0..124 (4 x per thread)

    v8f acc[2][2] = {};

    const float* zb = zp + (size_t)b * C_ * TY;
    const float* ab = Ap + (size_t)b * K2 * TX;

    // ---- prologue: stage chunk 0 into buffer 0 ----
    {
        ZStage z = load_z(zb, 0, lkp, y0 + ly);
        MStage m = load_m(ab, 0, lkp, x0 + lx);
        v2f t;
        t.x = z.r0.x; t.y = z.r1.x; Zs[0][lkp][ly + 0] = t;
        t.x = z.r0.y; t.y = z.r1.y; Zs[0][lkp][ly + 1] = t;
        t.x = m.q0.x; t.y = m.q1.x; Ms[0][lkp][lx + 0] = t;
        t.x = m.q0.y; t.y = m.q1.y; Ms[0][lkp][lx + 1] = t;
        t.x = m.q0.z; t.y = m.q1.z; Ms[0][lkp][lx + 2] = t;
        t.x = m.q0.w; t.y = m.q1.w; Ms[0][lkp][lx + 3] = t;
    }
    __syncthreads();

    for (int ch = 0; ch < NCHUNK; ++ch) {
        const int cur = ch & 1;
        const bool have_next = (ch + 1 < NCHUNK);

        // issue next chunk's global loads early (latency hides under WMMA)
        ZStage nz; MStage nm;
        if (have_next) {
            nz = load_z(zb, (ch + 1) * KC, lkp, y0 + ly);
            nm = load_m(ab, (ch + 1) * KC, lkp, x0 + lx);
        }

        // compute on current buffer: each fragment is one ds_load_b64
        #pragma unroll
        for (int ks = 0; ks < KC; ks += 4) {
            const int kp = (ks + kb) >> 1;          // even start -> exact pair
            v2f a0 = Zs[cur][kp][wm * 32      + l16];
            v2f a1 = Zs[cur][kp][wm * 32 + 16 + l16];
            v2f b0 = Ms[cur][kp][wn * 32      + l16];
            v2f b1 = Ms[cur][kp][wn * 32 + 16 + l16];
            acc[0][0] = wmma4(a0, b0, acc[0][0]);
            acc[0][1] = wmma4(a0, b1, acc[0][1]);
            acc[1][0] = wmma4(a1, b0, acc[1][0]);
            acc[1][1] = wmma4(a1, b1, acc[1][1]);
        }

        if (have_next) {
            const int nxt = cur ^ 1;
            v2f t;
            t.x = nz.r0.x; t.y = nz.r1.x; Zs[nxt][lkp][ly + 0] = t;
            t.x = nz.r0.y; t.y = nz.r1.y; Zs[nxt][lkp][ly + 1] = t;
            t.x = nm.q0.x; t.y = nm.q1.x; Ms[nxt][lkp][lx + 0] = t;
            t.x = nm.q0.y; t.y = nm.q1.y; Ms[nxt][lkp][lx + 1] = t;
            t.x = nm.q0.z; t.y = nm.q1.z; Ms[nxt][lkp][lx + 2] = t;
            t.x = nm.q0.w; t.y = nm.q1.w; Ms[nxt][lkp][lx + 3] = t;
        }
        __syncthreads();
    }

    int xl = min(max(xlen[b], 2), TX);
    int yl = max(min(max(ylen[b], 2), TY), xl);
    float* outb = lp_t + (size_t)b * TY * TX;
    #pragma unroll
    for (int mt = 0; mt < 2; ++mt) {
        #pragma unroll
        for (int nt = 0; nt < 2; ++nt) {
            int xg = x0 + wn * 32 + nt * 16 + l16;
            float bv = bias[b * TX + xg];
            #pragma unroll
            for (int r = 0; r < 8; ++r) {
                int yg = y0 + wm * 32 + mt * 16 + r + half * 8;
                float val = acc[mt][nt][r] + bv;
                bool ok = (xg < xl) && (yg < yl);
                outb[(size_t)yg * TX + xg] = ok ? val : NEGV;
            }
        }
    }
}

// ---------------------------------------------------------------------------
// Stage 3: forward DP.  One block per batch, thread = x.
// v[x,y] = lp[x,y] + max(v[x,y-1], v[x-1,y-1]); dirs bit = (diag strictly >).
// ---------------------------------------------------------------------------
__global__ __launch_bounds__(512) void mas_forward_kernel(
    const float* __restrict__ lp_t, unsigned int* __restrict__ dirs)
{
    __shared__ float sv[TX];
    const int b = blockIdx.x;
    const int x = threadIdx.x;
    const int lane = x & 31, wid = x >> 5;
    const float* lpb = lp_t + (size_t)b * TY * TX;
    unsigned int* db = dirs + (size_t)b * TY * (TX / 32);

    float v = (x == 0) ? lpb[0] : NEGV;
    sv[x] = v;
    if (x < TX / 32) db[x] = 0u;     // dirs row y=0 is all zeros
    __syncthreads();

    for (int y = 1; y < TY; ++y) {
        float lpv = lpb[(size_t)y * TX + x];
        float vm1 = (x == 0) ? NEGV : sv[x - 1];
        bool take = vm1 > v;
        v = lpv + fmaxf(v, vm1);
        unsigned long long bal = __ballot(take ? 1 : 0);  // wave32: low 32 bits
        __syncthreads();
        sv[x] = v;
        if (lane == 0) db[(size_t)y * (TX / 32) + wid] = (unsigned int)bal;
        __syncthreads();
    }
}

// ---------------------------------------------------------------------------
// Stage 4: backtrack per batch; dirs plane (128KB) cached in LDS.
// ---------------------------------------------------------------------------
#define SMEM_BT (TY * (TX / 32) * 4 + TY * 4 + TX * 4)   // 141312 B < 320KB

__global__ __launch_bounds__(256) void mas_backtrack_kernel(
    const unsigned int* __restrict__ dirs, const int* __restrict__ xlen,
    const int* __restrict__ ylen, int* __restrict__ xof,
    float* __restrict__ dur)
{
    extern __shared__ unsigned char smem[];
    unsigned int* sd = (unsigned int*)smem;                       // TY*16
    int* sx = (int*)(smem + (size_t)TY * (TX / 32) * 4);          // TY
    int* sc = (int*)(smem + (size_t)TY * (TX / 32) * 4 + TY * 4); // TX

    const int b = blockIdx.x, tid = threadIdx.x;
    const unsigned int* db = dirs + (size_t)b * TY * (TX / 32);
    for (int i = tid; i < TY * (TX / 32); i += 256) sd[i] = db[i];
    for (int i = tid; i < TX; i += 256) sc[i] = 0;
    __syncthreads();

    if (tid == 0) {
        int xl = min(max(xlen[b], 2), TX);
        int yl = max(min(max(ylen[b], 2), TY), xl);
        int idx = xl - 1;
        for (int y = TY - 1; y >= 0; --y) {
            if (y < yl) {
                sx[y] = idx;
                int d = (int)((sd[y * (TX / 32) + (idx >> 5)] >> (idx & 31)) & 1u);
                idx -= d;
            } else {
                sx[y] = -1;
            }
        }
    }
    __syncthreads();

    for (int y = tid; y < TY; y += 256) {
        int xo = sx[y];
        if (xo >= 0) atomicAdd(&sc[xo], 1);
        xof[(size_t)b * TY + y] = xo;
    }
    __syncthreads();
    for (int x = tid; x < TX; x += 256) dur[b * TX + x] = (float)sc[x];
}

// ---------------------------------------------------------------------------
// Stage 5a: attn[b][x][y] one-hot (overwrites the lp_t scratch region).
// ---------------------------------------------------------------------------
__global__ __launch_bounds__(256) void fill_attn_kernel(
    const int* __restrict__ xof, float* __restrict__ attn)
{
    size_t i4 = ((size_t)blockIdx.x * 256 + threadIdx.x) * 4;
    if (i4 >= (size_t)B_ * TX * TY) return;
    int y = (int)(i4 % TY);
    size_t bx = i4 / TY;
    int x = (int)(bx % TX);
    int b = (int)(bx / TX);
    const int* xo = xof + (size_t)b * TY + y;
    float4 o;
    o.x = (xo[0] == x) ? 1.0f : 0.0f;
    o.y = (xo[1] == x) ? 1.0f : 0.0f;
    o.z = (xo[2] == x) ? 1.0f : 0.0f;
    o.w = (xo[3] == x) ? 1.0f : 0.0f;
    *(float4*)(attn + i4) = o;
}

// Stage 5b: m_p_dur / logs_p_dur via one-hot gather along x_of.
__global__ __launch_bounds__(256) void fill_expand_kernel(
    const int* __restrict__ xof, const float* __restrict__ mp,
    const float* __restrict__ ls, float* __restrict__ mpd,
    float* __restrict__ lpd)
{
    size_t i4 = ((size_t)blockIdx.x * 256 + threadIdx.x) * 4;
    if (i4 >= (size_t)B_ * C_ * TY) return;
    int y = (int)(i4 % TY);
    size_t bc = i4 / TY;
    int b = (int)(bc / C_);
    const int* xo = xof + (size_t)b * TY + y;
    const float* mrow = mp + bc * TX;
    const float* lrow = ls + bc * TX;
    float4 mo, lo;
    {
        int x0v = xo[0]; mo.x = (x0v >= 0) ? mrow[x0v] : 0.f; lo.x = (x0v >= 0) ? lrow[x0v] : 0.f;
        int x1v = xo[1]; mo.y = (x1v >= 0) ? mrow[x1v] : 0.f; lo.y = (x1v >= 0) ? lrow[x1v] : 0.f;
        int x2v = xo[2]; mo.z = (x2v >= 0) ? mrow[x2v] : 0.f; lo.z = (x2v >= 0) ? lrow[x2v] : 0.f;
        int x3v = xo[3]; mo.w = (x3v >= 0) ? mrow[x3v] : 0.f; lo.w = (x3v >= 0) ? lrow[x3v] : 0.f;
    }
    *(float4*)(mpd + i4) = mo;
    *(float4*)(lpd + i4) = lo;
}

// ---------------------------------------------------------------------------
extern "C" void kernel_launch(void* const* d_in, const int* in_sizes, int n_in,
                              void* d_out, int out_size, void* d_ws, size_t ws_size,
                              hipStream_t stream) {
    (void)in_sizes; (void)n_in; (void)out_size; (void)ws_size;

    const float* z_p    = (const float*)d_in[0];   // [B,C,Ty]
    const float* m_p    = (const float*)d_in[1];   // [B,C,Tx]
    const float* logs_p = (const float*)d_in[2];   // [B,C,Tx]
    const int*   x_len  = (const int*)d_in[3];     // [B]
    const int*   y_len  = (const int*)d_in[4];     // [B]

    float* out  = (float*)d_out;
    float* dur  = out;                                      // [B,1,Tx]
    float* attn = out + (size_t)B_ * TX;                    // [B,1,Tx,Ty] (lp_t scratch first)
    float* mpd  = attn + (size_t)B_ * TX * TY;              // [B,C,Ty]
    float* lpd  = mpd + (size_t)B_ * C_ * TY;               // [B,C,Ty]

    // workspace: A^T (25.2MB) | bias (64KB) | dirs (4MB) | x_of (256KB)
    float* Ap   = (float*)d_ws;
    float* bias = Ap + (size_t)B_ * K2 * TX;
    unsigned int* dirs = (unsigned int*)(bias + (size_t)B_ * TX);
    int* xof = (int*)(dirs + (size_t)B_ * TY * (TX / 32));

    prep_a_kernel<<<(B_ * C_ * TX) / 256, 256, 0, stream>>>(m_p, logs_p, Ap);
    prep_bias_kernel<<<(B_ * TX) / 256, 256, 0, stream>>>(m_p, logs_p, bias);

    dim3 ggrid(TX / BTN, TY / BTM, B_);                     // (4, 32, 32)
    gemm_logp_kernel<<<ggrid, 256, 0, stream>>>(z_p, Ap, bias, x_len, y_len, attn);

    mas_forward_kernel<<<B_, TX, 0, stream>>>(attn, dirs);

    hipFuncSetAttribute((const void*)mas_backtrack_kernel,
                        hipFuncAttributeMaxDynamicSharedMemorySize, SMEM_BT);
    mas_backtrack_kernel<<<B_, 256, SMEM_BT, stream>>>(dirs, x_len, y_len, xof, dur);

    fill_attn_kernel<<<((size_t)B_ * TX * TY) / 1024, 256, 0, stream>>>(xof, attn);
    fill_expand_kernel<<<((size_t)B_ * C_ * TY) / 1024, 256, 0, stream>>>(
        xof, m_p, logs_p, mpd, lpd);
}